// SoftQNetwork_50517405335739
// MI455X (gfx1250) — compile-verified
//
#include <hip/hip_runtime.h>

typedef __attribute__((ext_vector_type(2))) float v2f;
typedef __attribute__((ext_vector_type(8))) float v8f;
typedef __attribute__((ext_vector_type(4))) unsigned int v4u;
typedef __attribute__((ext_vector_type(8))) int v8i;
typedef __attribute__((ext_vector_type(4))) int v4i;

#define OBS   64
#define ACTD  16
#define DDIM  80      // OBS + ACT
#define HID   256
#define MIX   16
#define FDIM  4096    // HID * MIX
#define TOPK  4
#define KCH   128     // K-chunk width for GEMM2
#define LN_EPS 1e-5f

static __device__ __forceinline__ v8f wmma_f32(v2f a, v2f b, v8f c) {
  // D = A(16x4 f32) * B(4x16 f32) + C(16x16 f32)
  return __builtin_amdgcn_wmma_f32_16x16x4_f32(
      /*neg_a=*/false, a, /*neg_b=*/false, b,
      /*c_mod=*/(short)0, c, /*reuse_a=*/false, /*reuse_b=*/false);
}

// TDM: async DMA of a 16-row x 128-col f32 tile (row stride 4096 elements)
// from global into LDS, padding +4 DWORDs after every 128 DWORDs so rows land
// at a 132-float stride (bank-conflict-free for the A-fragment reads).
static __device__ __forceinline__ void tdm_load_tile16x128(
    const float* gsrc, unsigned lds_byte_off)
{
  const unsigned long long ga = (unsigned long long)(uintptr_t)gsrc;
  v4u g0;
  g0.x = 1u;                                   // count=1 (valid), gather off
  g0.y = lds_byte_off;                         // lds_addr (bytes)
  g0.z = (unsigned)ga;                         // global_addr[31:0]
  g0.w = ((unsigned)(ga >> 32) & 0x1FFFFFFu)   // global_addr[56:32]
       | (2u << 30);                           // type = 2 ("image")
  v8i g1;
  g1[0] = (2 << 16)    // data_size = 4 bytes
        | (1 << 20)    // pad_enable
        | (6 << 22)    // pad_interval: 2^(6+1) = 128 DWORDs
        | (3 << 25);   // pad_amount: 3+1 = 4 DWORDs
  g1[1] = (int)(4096u << 16);   // tensor_dim0[15:0] = 4096 (atomic_barrier_addr = 0)
  g1[2] = (16 << 16);           // tensor_dim0[31:16]=0 | tensor_dim1[15:0] = 16
  g1[3] = (KCH << 16);          // tensor_dim1[31:16]=0 | tile_dim0 = 128
  g1[4] = 16;                   // tile_dim1 = 16, tile_dim2 = 0
  g1[5] = 4096;                 // tensor_dim0_stride[31:0] = 4096 elements
  g1[6] = 0;
  g1[7] = 0;
  v4i z4 = {0, 0, 0, 0};                    // groups 2/3 unused (2-D tensor)
  v8i z8 = {0, 0, 0, 0, 0, 0, 0, 0};        // extra group (clang-23 6-arg form)
  __builtin_amdgcn_tensor_load_to_lds(g0, g1, z4, z4, z8, 0);
}

// ---------------------------------------------------------------------------
// Kernel 1: gating — softmax over 16 experts, top-4, renormalize, scatter.
// ---------------------------------------------------------------------------
__global__ void __launch_bounds__(256) gate_scores_kernel(
    const float* __restrict__ x, const float* __restrict__ a,
    const float* __restrict__ Wg, const float* __restrict__ bg,
    float* __restrict__ scores, int B)
{
  const int r = blockIdx.x * 256 + threadIdx.x;
  if (r >= B) return;

  float xa[DDIM];
  #pragma unroll
  for (int k = 0; k < OBS; ++k)  xa[k]       = x[r * OBS + k];
  #pragma unroll
  for (int k = 0; k < ACTD; ++k) xa[OBS + k] = a[r * ACTD + k];

  float g[MIX];
  #pragma unroll
  for (int j = 0; j < MIX; ++j) {
    float s = bg[j];
    const float* wrow = Wg + j * DDIM;
    #pragma unroll
    for (int k = 0; k < DDIM; ++k) s += xa[k] * wrow[k];
    g[j] = s;
  }

  float mx = g[0];
  #pragma unroll
  for (int j = 1; j < MIX; ++j) mx = fmaxf(mx, g[j]);
  float sum = 0.f;
  #pragma unroll
  for (int j = 0; j < MIX; ++j) { g[j] = expf(g[j] - mx); sum += g[j]; }
  const float inv = 1.f / sum;
  #pragma unroll
  for (int j = 0; j < MIX; ++j) g[j] *= inv;

  bool used[MIX];
  #pragma unroll
  for (int j = 0; j < MIX; ++j) used[j] = false;
  float tsum = 0.f;
  #pragma unroll
  for (int t = 0; t < TOPK; ++t) {
    int best = 0; float bv = -1.f;   // softmax probs >= 0
    #pragma unroll
    for (int j = 0; j < MIX; ++j)
      if (!used[j] && g[j] > bv) { bv = g[j]; best = j; }  // first-max tiebreak
    used[best] = true; tsum += bv;
  }
  const float invt = 1.f / tsum;
  #pragma unroll
  for (int j = 0; j < MIX; ++j)
    scores[r * MIX + j] = used[j] ? g[j] * invt : 0.f;
}

// ---------------------------------------------------------------------------
// Kernel 2: h1 = relu(LN(xa @ W1^T + b1))   [B,4096]  via f32 WMMA
// Block = 512 threads (16 waves): 16 batch rows x full 4096 cols.
// ---------------------------------------------------------------------------
__global__ void __launch_bounds__(512) gemm1_ln_relu_kernel(
    const float* __restrict__ x, const float* __restrict__ a,
    const float* __restrict__ W1, const float* __restrict__ b1,
    const float* __restrict__ g1, const float* __restrict__ be1,
    float* __restrict__ h1)
{
  __shared__ float lA[16][DDIM + 4];
  __shared__ float redS[16][16];
  __shared__ float redQ[16][16];
  __shared__ float rmean[16];
  __shared__ float rrstd[16];

  const int tid  = threadIdx.x;
  const int wave = tid >> 5;
  const int lane = tid & 31;
  const int half = lane >> 4;
  const int l16  = lane & 15;
  const int row0 = blockIdx.x << 4;
  const int nbase = wave << 8;

  for (int idx = tid; idx < 16 * DDIM; idx += 512) {
    const int r = idx / DDIM, c = idx - r * DDIM;
    const int gr = row0 + r;
    lA[r][c] = (c < OBS) ? x[gr * OBS + c] : a[gr * ACTD + (c - OBS)];
  }
  __syncthreads();

  v8f acc[16] = {};
  for (int k = 0; k < DDIM; k += 4) {
    v2f af;
    af.x = lA[l16][k + 2 * half];
    af.y = lA[l16][k + 2 * half + 1];
    #pragma unroll
    for (int t = 0; t < 16; ++t) {
      const int n = nbase + t * 16 + l16;
      const v2f bf = *(const v2f*)(W1 + n * DDIM + k + 2 * half);
      acc[t] = wmma_f32(af, bf, acc[t]);
    }
  }

  float s[8], q[8];
  #pragma unroll
  for (int i = 0; i < 8; ++i) { s[i] = 0.f; q[i] = 0.f; }
  #pragma unroll
  for (int t = 0; t < 16; ++t) {
    const float bv = b1[nbase + t * 16 + l16];
    #pragma unroll
    for (int i = 0; i < 8; ++i) {
      const float v = acc[t][i] + bv;
      acc[t][i] = v;
      s[i] += v; q[i] += v * v;
    }
  }
  #pragma unroll
  for (int i = 0; i < 8; ++i) {
    #pragma unroll
    for (int m = 1; m < 16; m <<= 1) {
      s[i] += __shfl_xor(s[i], m, 32);
      q[i] += __shfl_xor(q[i], m, 32);
    }
  }
  if (l16 == 0) {
    #pragma unroll
    for (int i = 0; i < 8; ++i) {
      redS[wave][half * 8 + i] = s[i];
      redQ[wave][half * 8 + i] = q[i];
    }
  }
  __syncthreads();
  if (tid < 16) {   // deterministic fixed-order cross-wave sum
    float ss = 0.f, qq = 0.f;
    #pragma unroll
    for (int w = 0; w < 16; ++w) { ss += redS[w][tid]; qq += redQ[w][tid]; }
    const float mean = ss * (1.0f / FDIM);
    const float var  = qq * (1.0f / FDIM) - mean * mean;
    rmean[tid] = mean;
    rrstd[tid] = rsqrtf(var + LN_EPS);
  }
  __syncthreads();

  #pragma unroll
  for (int i = 0; i < 8; ++i) {
    const int r = half * 8 + i;
    const float mean = rmean[r];
    const float rstd = rrstd[r];
    #pragma unroll
    for (int t = 0; t < 16; ++t) {
      const int n = nbase + t * 16 + l16;
      float v = (acc[t][i] - mean) * rstd * g1[n] + be1[n];
      v = fmaxf(v, 0.0f);
      h1[(size_t)(row0 + r) * FDIM + n] = v;
    }
  }
}

// ---------------------------------------------------------------------------
// Kernel 3: z = LN(h1 @ W2^T + b2); v = relu(z);
//           out[b] = b3 + (1/16) * sum_n v[n] * scores[b][n%16] * W3[n/16]
// A-chunks (16x128) staged by the Tensor Data Mover into a double-buffered
// padded LDS tile, overlapping the WMMA sweep of the previous chunk.
// ---------------------------------------------------------------------------
__global__ void __launch_bounds__(512) gemm2_ln_mix_kernel(
    const float* __restrict__ h1, const float* __restrict__ W2,
    const float* __restrict__ b2, const float* __restrict__ g2,
    const float* __restrict__ be2, const float* __restrict__ W3,
    const float* __restrict__ b3, const float* __restrict__ scores,
    float* __restrict__ out)
{
  __shared__ float lA[2][16][KCH + 4];   // +4: TDM pad (4 DW per 128 DW)
  __shared__ float redS[16][16];
  __shared__ float redQ[16][16];
  __shared__ float rmean[16];
  __shared__ float rrstd[16];
  __shared__ float lsc[16][MIX];
  __shared__ float lW3[HID];

  const int tid  = threadIdx.x;
  const int wave = tid >> 5;
  const int lane = tid & 31;
  const int half = lane >> 4;
  const int l16  = lane & 15;
  const int row0 = blockIdx.x << 4;
  const int nbase = wave << 8;

  if (tid < HID) lW3[tid] = W3[tid];
  if (tid >= HID && tid < HID + 16 * MIX) {
    const int idx = tid - HID;
    lsc[idx >> 4][idx & 15] = scores[(row0 + (idx >> 4)) * MIX + (idx & 15)];
  }

  const unsigned ldsA0 = (unsigned)(size_t)(&lA[0][0][0]);
  const unsigned ldsA1 = (unsigned)(size_t)(&lA[1][0][0]);
  const float* abase = h1 + (size_t)row0 * FDIM;

  // Prime the pipeline: TDM fetch of chunk 0 into buffer 0 (wave 0 only;
  // TDM issues once per wave, EXEC-independent).
  if (wave == 0) tdm_load_tile16x128(abase, ldsA0);

  v8f acc[16] = {};
  for (int c = 0; c < FDIM / KCH; ++c) {
    const int buf = c & 1;
    __builtin_amdgcn_s_wait_tensorcnt(0);   // wave0 blocks until chunk c landed
    __syncthreads();                        // publish LDS tile to all waves
    // Overlap: fetch chunk c+1 into the other buffer while we compute on c.
    if (wave == 0 && (c + 1) < FDIM / KCH)
      tdm_load_tile16x128(abase + (size_t)(c + 1) * KCH, buf ? ldsA0 : ldsA1);

    const size_t kc = (size_t)c * KCH;
    for (int k = 0; k < KCH; k += 4) {
      v2f af;
      af.x = lA[buf][l16][k + 2 * half];
      af.y = lA[buf][l16][k + 2 * half + 1];
      #pragma unroll
      for (int t = 0; t < 16; ++t) {
        const int n = nbase + t * 16 + l16;
        const v2f bf = *(const v2f*)(W2 + (size_t)n * FDIM + kc + k + 2 * half);
        acc[t] = wmma_f32(af, bf, acc[t]);
      }
    }
  }

  // bias + LN statistics
  float s[8], q[8];
  #pragma unroll
  for (int i = 0; i < 8; ++i) { s[i] = 0.f; q[i] = 0.f; }
  #pragma unroll
  for (int t = 0; t < 16; ++t) {
    const float bv = b2[nbase + t * 16 + l16];
    #pragma unroll
    for (int i = 0; i < 8; ++i) {
      const float v = acc[t][i] + bv;
      acc[t][i] = v;
      s[i] += v; q[i] += v * v;
    }
  }
  #pragma unroll
  for (int i = 0; i < 8; ++i) {
    #pragma unroll
    for (int m = 1; m < 16; m <<= 1) {
      s[i] += __shfl_xor(s[i], m, 32);
      q[i] += __shfl_xor(q[i], m, 32);
    }
  }
  if (l16 == 0) {
    #pragma unroll
    for (int i = 0; i < 8; ++i) {
      redS[wave][half * 8 + i] = s[i];
      redQ[wave][half * 8 + i] = q[i];
    }
  }
  __syncthreads();
  if (tid < 16) {
    float ss = 0.f, qq = 0.f;
    #pragma unroll
    for (int w = 0; w < 16; ++w) { ss += redS[w][tid]; qq += redQ[w][tid]; }
    const float mean = ss * (1.0f / FDIM);
    const float var  = qq * (1.0f / FDIM) - mean * mean;
    rmean[tid] = mean;
    rrstd[tid] = rsqrtf(var + LN_EPS);
  }
  __syncthreads();

  // LN + relu + expert-mixture + W3 projection, fused (no h2 buffer)
  float p[8];
  #pragma unroll
  for (int i = 0; i < 8; ++i) p[i] = 0.f;
  #pragma unroll
  for (int t = 0; t < 16; ++t) {
    const int n = nbase + t * 16 + l16;
    const float gn = g2[n], bn = be2[n];
    const float w3 = lW3[n >> 4];
    #pragma unroll
    for (int i = 0; i < 8; ++i) {
      const int r = half * 8 + i;
      float v = (acc[t][i] - rmean[r]) * rrstd[r] * gn + bn;
      v = fmaxf(v, 0.0f);
      p[i] += v * lsc[r][n & 15] * w3;
    }
  }
  #pragma unroll
  for (int i = 0; i < 8; ++i) {
    #pragma unroll
    for (int m = 1; m < 16; m <<= 1) p[i] += __shfl_xor(p[i], m, 32);
  }
  if (l16 == 0) {
    #pragma unroll
    for (int i = 0; i < 8; ++i) redS[wave][half * 8 + i] = p[i];  // reuse redS
  }
  __syncthreads();
  if (tid < 16) {
    float y = 0.f;
    #pragma unroll
    for (int w = 0; w < 16; ++w) y += redS[w][tid];
    out[row0 + tid] = y * (1.0f / MIX) + b3[0];
  }
}

// ---------------------------------------------------------------------------
extern "C" void kernel_launch(void* const* d_in, const int* in_sizes, int n_in,
                              void* d_out, int out_size, void* d_ws, size_t ws_size,
                              hipStream_t stream) {
  const float* x   = (const float*)d_in[0];
  const float* a   = (const float*)d_in[1];
  const float* W1  = (const float*)d_in[2];
  const float* b1  = (const float*)d_in[3];
  const float* W2  = (const float*)d_in[4];
  const float* b2  = (const float*)d_in[5];
  const float* W3  = (const float*)d_in[6];
  const float* b3  = (const float*)d_in[7];
  const float* g1  = (const float*)d_in[8];
  const float* be1 = (const float*)d_in[9];
  const float* g2  = (const float*)d_in[10];
  const float* be2 = (const float*)d_in[11];
  const float* Wg  = (const float*)d_in[12];
  const float* bg  = (const float*)d_in[13];
  float* out = (float*)d_out;

  const int B = in_sizes[0] / OBS;   // 16384

  // workspace: scores [B,16] f32, then h1 [B,4096] f32
  float* scores = (float*)d_ws;
  float* h1     = scores + (size_t)B * MIX;

  gate_scores_kernel<<<(B + 255) / 256, 256, 0, stream>>>(x, a, Wg, bg, scores, B);
  gemm1_ln_relu_kernel<<<B / 16, 512, 0, stream>>>(x, a, W1, b1, g1, be1, h1);
  gemm2_ln_mix_kernel<<<B / 16, 512, 0, stream>>>(h1, W2, b2, g2, be2, W3, b3, scores, out);
}